// MultiHead_32298154066764
// MI455X (gfx1250) — compile-verified
//
#include <hip/hip_runtime.h>
#include <hip/hip_bf16.h>

typedef __attribute__((ext_vector_type(2))) float v2f;
typedef __attribute__((ext_vector_type(8))) float v8f;

#define EPSF 1e-5f

__device__ __forceinline__ v8f wmma4(v2f a, v2f b, v8f c) {
    // D = A(16x4 f32) x B(4x16 f32) + C(16x16 f32)
    return __builtin_amdgcn_wmma_f32_16x16x4_f32(
        /*neg_a=*/false, a, /*neg_b=*/false, b,
        /*c_mod=*/(short)0, c, /*reuse_a=*/false, /*reuse_b=*/false);
}

// -------------------------------------------------------------------------
// Pack W[K,D] (row-major) into WMMA-B pair layout:
// WP[kp*D + c] = (W[2kp][c], W[2kp+1][c])  -> one b64 load per lane per step.
// -------------------------------------------------------------------------
__global__ void pack_wpair_kernel(const float* __restrict__ W,
                                  v2f* __restrict__ WP, int K, int D) {
    int t = blockIdx.x * blockDim.x + threadIdx.x;
    int total = (K >> 1) * D;
    if (t >= total) return;
    int kp = t / D, c = t - kp * D;
    v2f p;
    p.x = W[(size_t)(2 * kp) * D + c];
    p.y = W[(size_t)(2 * kp + 1) * D + c];
    WP[t] = p;
}

// w2 [128,17] -> padded pair layout [64][32]; b2 -> padded [32]
__global__ void pack_w2_kernel(const float* __restrict__ w2,
                               const float* __restrict__ b2,
                               v2f* __restrict__ w2p,
                               float* __restrict__ b2p) {
    int t = blockIdx.x * blockDim.x + threadIdx.x;
    if (t < 64 * 32) {
        int kp = t >> 5, c = t & 31;
        v2f p;
        p.x = (c < 17) ? w2[(2 * kp) * 17 + c] : 0.f;
        p.y = (c < 17) ? w2[(2 * kp + 1) * 17 + c] : 0.f;
        w2p[t] = p;
    }
    if (t < 32) b2p[t] = (t < 17) ? b2[t] : 0.f;
}

// -------------------------------------------------------------------------
// fp32 WMMA GEMM, compile-time K/D: Y[:,0..Dstore) = X[N,K] @ W[K,D] + bias.
// One 16x16 output tile per wave; 8 waves / 256-thread block. Fully unrolled
// K-loop: all load offsets are immediates off two base pointers.
// -------------------------------------------------------------------------
template <int K, int D>
__global__ void gemm_f32_kernel(const float* __restrict__ X,
                                const v2f* __restrict__ WP,
                                const float* __restrict__ bias,
                                float* __restrict__ Y,
                                int ldY, int Dstore, int nTiles) {
    int wave = threadIdx.x >> 5;
    int lane = threadIdx.x & 31;
    int tile = blockIdx.x * 8 + wave;
    if (tile >= nTiles) return;
    constexpr int tilesD = D >> 4;
    int tn = tile / tilesD, td = tile - tn * tilesD;
    int r0 = tn << 4, c0 = td << 4;
    int hl   = lane & 15;
    int kOff = (lane >> 4) << 1;              // lanes 0-15: K=0,1; 16-31: K=2,3
    int row  = r0 + hl;
    int col  = c0 + hl;
    const float* xr = X + (size_t)row * K + kOff;
    const v2f*   wp = WP + (size_t)(kOff >> 1) * D + col;
    v8f acc = {0.f, 0.f, 0.f, 0.f, 0.f, 0.f, 0.f, 0.f};
#pragma unroll
    for (int k = 0; k < K; k += 4) {
        v2f a = *(const v2f*)(xr + k);        // aligned: row*K + kOff + k is even
        v2f b = wp[(k >> 1) * D];
        acc = wmma4(a, b, acc);
    }
    if (col < Dstore) {
        float bv = bias[col];
        int rbase = r0 + ((lane >> 4) << 3);  // lanes 0-15: rows 0-7; 16-31: rows 8-15
#pragma unroll
        for (int r = 0; r < 8; ++r)
            Y[(size_t)(rbase + r) * ldY + col] = acc[r] + bv;
    }
}

// -------------------------------------------------------------------------
// Classifier GEMM1 with fused NN gather: K=256 split into 4 x 64-wide blocks
// taken from f0 (identity) and f1/f2/f3 (indirect via idx arrays). W packed.
// -------------------------------------------------------------------------
__global__ void gemm_cat_kernel(const float* __restrict__ F0,
                                const float* __restrict__ F1,
                                const float* __restrict__ F2,
                                const float* __restrict__ F3,
                                const int* __restrict__ I1,
                                const int* __restrict__ I2,
                                const int* __restrict__ I3,
                                const v2f* __restrict__ WP,
                                const float* __restrict__ bias,
                                float* __restrict__ H, int nTiles) {
    constexpr int D = 128;
    int wave = threadIdx.x >> 5;
    int lane = threadIdx.x & 31;
    int tile = blockIdx.x * 8 + wave;
    if (tile >= nTiles) return;
    int tn = tile >> 3, td = tile & 7;
    int r0 = tn << 4, c0 = td << 4;
    int hl   = lane & 15;
    int kOff = (lane >> 4) << 1;
    int row  = r0 + hl;
    int col  = c0 + hl;
    const float* pa0 = F0 + (size_t)row * 64 + kOff;
    const float* pa1 = F1 + (size_t)I1[row] * 64 + kOff;
    const float* pa2 = F2 + (size_t)I2[row] * 64 + kOff;
    const float* pa3 = F3 + (size_t)I3[row] * 64 + kOff;
    const v2f* wp = WP + (size_t)(kOff >> 1) * D + col;
    v8f acc = {0.f, 0.f, 0.f, 0.f, 0.f, 0.f, 0.f, 0.f};
#pragma unroll
    for (int j = 0; j < 4; ++j) {
        const float* xr = (j == 0) ? pa0 : (j == 1) ? pa1 : (j == 2) ? pa2 : pa3;
#pragma unroll
        for (int kk = 0; kk < 64; kk += 4) {
            v2f a = *(const v2f*)(xr + kk);
            v2f b = wp[((j * 64 + kk) >> 1) * D];
            acc = wmma4(a, b, acc);
        }
    }
    float bv = bias[col];
    int rbase = r0 + ((lane >> 4) << 3);
#pragma unroll
    for (int r = 0; r < 8; ++r)
        H[(size_t)(rbase + r) * D + col] = acc[r] + bv;
}

// -------------------------------------------------------------------------
// Column sum / sum-of-squares reduction for training-mode BatchNorm.
// stats layout: [0..D) = sum, [D..2D) = sumsq. Must be pre-zeroed.
// -------------------------------------------------------------------------
__global__ void colstats_kernel(const float* __restrict__ Y,
                                float* __restrict__ stats, int N, int D) {
    __shared__ float s_sum[256];
    __shared__ float s_sq[256];
    int t   = threadIdx.x;
    int c   = t % D;
    int rl  = t / D;
    int rpb = 256 / D;
    float sum = 0.f, sq = 0.f;
    for (int r = blockIdx.x * rpb + rl; r < N; r += gridDim.x * rpb) {
        float v = Y[(size_t)r * D + c];
        sum += v;
        sq  += v * v;
    }
    s_sum[t] = sum;
    s_sq[t]  = sq;
    __syncthreads();
    if (rl == 0) {
        for (int i = 1; i < rpb; ++i) {
            sum += s_sum[c + i * D];
            sq  += s_sq[c + i * D];
        }
        atomicAdd(&stats[c], sum);
        atomicAdd(&stats[D + c], sq);
    }
}

__global__ void bn_finalize_kernel(const float* __restrict__ stats,
                                   const float* __restrict__ g,
                                   const float* __restrict__ beta,
                                   float* __restrict__ ss, int N, int D) {
    int c = blockIdx.x * blockDim.x + threadIdx.x;
    if (c >= D) return;
    float invN  = 1.0f / (float)N;
    float mean  = stats[c] * invN;
    float var   = stats[D + c] * invN - mean * mean;
    float scale = g[c] * rsqrtf(var + EPSF);
    ss[c]     = scale;
    ss[D + c] = beta[c] - mean * scale;
}

__global__ void bn_relu_kernel(float* __restrict__ Y,
                               const float* __restrict__ ss,
                               size_t total, int D) {
    size_t i = (size_t)blockIdx.x * blockDim.x + threadIdx.x;
    if (i >= total) return;
    int c = (int)(i % (size_t)D);
    float v = Y[i] * ss[c] + ss[D + c];
    Y[i] = v > 0.f ? v : 0.f;
}

// -------------------------------------------------------------------------
// Candidate packing for WMMA kNN: clo = (x,y), chi = (z, -0.5*|c|^2).
// -------------------------------------------------------------------------
__global__ void pack_cand_kernel(const float* __restrict__ P,
                                 float* __restrict__ clo,
                                 float* __restrict__ chi, int N) {
    int i = blockIdx.x * blockDim.x + threadIdx.x;
    if (i >= N) return;
    float x = P[3 * i], y = P[3 * i + 1], z = P[3 * i + 2];
    clo[2 * i]     = x;
    clo[2 * i + 1] = y;
    chi[2 * i]     = z;
    chi[2 * i + 1] = -0.5f * (x * x + y * y + z * z);
}

// -------------------------------------------------------------------------
// k=1 NN argmin via WMMA f32 16x16x4. Score s = q.c - 0.5|c|^2 (4th K slot
// carries the bias: A row = (qx,qy,qz,1), B col = (cx,cy,cz,-0.5|c|^2)).
// argmin distance == argmax s. Each wave owns 32 queries (two A tiles) and
// streams candidate tiles; per-lane running (max, first-argmax) then a
// 16-lane shfl_xor reduction with first-index tie-break.
// -------------------------------------------------------------------------
__global__ void knn_argmin_kernel(const float* __restrict__ Q,
                                  const float* __restrict__ clo,
                                  const float* __restrict__ chi,
                                  int* __restrict__ idxOut, int NQ, int NC) {
    int wave = threadIdx.x >> 5;
    int lane = threadIdx.x & 31;
    int q0 = (blockIdx.x * 8 + wave) * 32;
    if (q0 >= NQ) return;
    int  hl = lane & 15;
    bool hi = lane >= 16;

    v2f a0, a1;
    {
        int r = q0 + hl;
        if (!hi) { a0.x = Q[3 * r];     a0.y = Q[3 * r + 1]; }
        else     { a0.x = Q[3 * r + 2]; a0.y = 1.0f; }
        r += 16;
        if (!hi) { a1.x = Q[3 * r];     a1.y = Q[3 * r + 1]; }
        else     { a1.x = Q[3 * r + 2]; a1.y = 1.0f; }
    }
    const v2f* src = (const v2f*)(hi ? chi : clo);

    float best0[8], best1[8];
    int   bi0[8], bi1[8];
#pragma unroll
    for (int r = 0; r < 8; ++r) {
        best0[r] = -3.4e38f; best1[r] = -3.4e38f;
        bi0[r] = 0; bi1[r] = 0;
    }
    const v8f zero = {0.f, 0.f, 0.f, 0.f, 0.f, 0.f, 0.f, 0.f};
#pragma unroll 2
    for (int jc = 0; jc < NC; jc += 16) {
        v2f b = src[jc + hl];                 // single aligned b64
        v8f s0 = wmma4(a0, b, zero);
        v8f s1 = wmma4(a1, b, zero);
        int myidx = jc + hl;                  // candidate column owned by lane
#pragma unroll
        for (int r = 0; r < 8; ++r) {
            if (s0[r] > best0[r]) { best0[r] = s0[r]; bi0[r] = myidx; }
            if (s1[r] > best1[r]) { best1[r] = s1[r]; bi1[r] = myidx; }
        }
    }
    // reduce across the 16 lanes holding the same row half
#pragma unroll
    for (int off = 8; off >= 1; off >>= 1) {
#pragma unroll
        for (int r = 0; r < 8; ++r) {
            float o0 = __shfl_xor(best0[r], off);
            int  oi0 = __shfl_xor(bi0[r], off);
            if (o0 > best0[r] || (o0 == best0[r] && oi0 < bi0[r])) {
                best0[r] = o0; bi0[r] = oi0;
            }
            float o1 = __shfl_xor(best1[r], off);
            int  oi1 = __shfl_xor(bi1[r], off);
            if (o1 > best1[r] || (o1 == best1[r] && oi1 < bi1[r])) {
                best1[r] = o1; bi1[r] = oi1;
            }
        }
    }
    if (hl == 0) {
        int rb0 = q0 + (hi ? 8 : 0);
        int rb1 = q0 + 16 + (hi ? 8 : 0);
#pragma unroll
        for (int r = 0; r < 8; ++r) {
            idxOut[rb0 + r] = bi0[r];
            idxOut[rb1 + r] = bi1[r];
        }
    }
}

__global__ void zero_kernel(float* __restrict__ p, int n) {
    int i = blockIdx.x * blockDim.x + threadIdx.x;
    if (i < n) p[i] = 0.f;
}

// -------------------------------------------------------------------------
extern "C" void kernel_launch(void* const* d_in, const int* in_sizes, int n_in,
                              void* d_out, int out_size, void* d_ws, size_t ws_size,
                              hipStream_t stream) {
    (void)in_sizes; (void)n_in; (void)out_size; (void)ws_size;
    static const int NS[4]    = {98304, 24576, 6144, 1536};
    static const int FDIMS[4] = {64, 128, 256, 512};

    const float* P[4] = {(const float*)d_in[0], (const float*)d_in[2],
                         (const float*)d_in[4], (const float*)d_in[6]};
    const float* X[4] = {(const float*)d_in[1], (const float*)d_in[3],
                         (const float*)d_in[5], (const float*)d_in[7]};
    const float* SW[4]  = {(const float*)d_in[8],  (const float*)d_in[9],
                           (const float*)d_in[10], (const float*)d_in[11]};
    const float* SB[4]  = {(const float*)d_in[12], (const float*)d_in[13],
                           (const float*)d_in[14], (const float*)d_in[15]};
    const float* SG[4]  = {(const float*)d_in[16], (const float*)d_in[17],
                           (const float*)d_in[18], (const float*)d_in[19]};
    const float* SBe[4] = {(const float*)d_in[20], (const float*)d_in[21],
                           (const float*)d_in[22], (const float*)d_in[23]};
    const float* w1    = (const float*)d_in[24];
    const float* b1    = (const float*)d_in[25];
    const float* g1    = (const float*)d_in[26];
    const float* beta1 = (const float*)d_in[27];
    const float* w2    = (const float*)d_in[28];
    const float* b2    = (const float*)d_in[29];
    float* out = (float*)d_out;

    // ---- workspace layout (floats) ----
    float* ws = (float*)d_ws;
    size_t o = 0;
    float* Y[4];
    for (int i = 0; i < 4; ++i) { Y[i] = ws + o; o += (size_t)NS[i] * 64; }
    float* H = ws + o; o += (size_t)NS[0] * 128;
    float *CLO[4], *CHI[4];
    for (int i = 1; i < 4; ++i) {
        CLO[i] = ws + o; o += (size_t)NS[i] * 2;
        CHI[i] = ws + o; o += (size_t)NS[i] * 2;
    }
    int* IDX[4];
    for (int i = 1; i < 4; ++i) { IDX[i] = (int*)(ws + o); o += (size_t)NS[0]; }
    v2f* WPS[4];   // stage weights in pair layout: (F_i/2)*64 pairs
    for (int i = 0; i < 4; ++i) { WPS[i] = (v2f*)(ws + o); o += (size_t)FDIMS[i] * 64; }
    v2f* WP1 = (v2f*)(ws + o); o += 256 * 128;   // w1 pairs
    v2f* W2P = (v2f*)(ws + o); o += 128 * 32;    // w2 padded pairs [64][32]
    float* B2P = ws + o; o += 32;
    float* STATS = ws + o; o += 768;   // 4 stages x 128  +  h x 256 (contiguous)
    float* SS    = ws + o; o += 768;   // scale/shift, same layout

    // stats must be zero every call (harness does not re-poison scratch)
    zero_kernel<<<3, 256, 0, stream>>>(STATS, 768);

    // ---- pack all weights into WMMA-B pair layout ----
    for (int i = 0; i < 4; ++i) {
        int total = (FDIMS[i] / 2) * 64;
        pack_wpair_kernel<<<(total + 255) / 256, 256, 0, stream>>>(
            SW[i], WPS[i], FDIMS[i], 64);
    }
    pack_wpair_kernel<<<(128 * 128 + 255) / 256, 256, 0, stream>>>(w1, WP1, 256, 128);
    pack_w2_kernel<<<8, 256, 0, stream>>>(w2, b2, W2P, B2P);

    // ---- per-stage MLP: y = x@w + b; BN stats; finalize; BN+ReLU in place ----
    for (int i = 0; i < 4; ++i) {
        int N = NS[i];
        int nTiles = (N / 16) * (64 / 16);
        int grid = (nTiles + 7) / 8;
        switch (FDIMS[i]) {
            case 64:
                gemm_f32_kernel<64, 64><<<grid, 256, 0, stream>>>(
                    X[i], WPS[i], SB[i], Y[i], 64, 64, nTiles); break;
            case 128:
                gemm_f32_kernel<128, 64><<<grid, 256, 0, stream>>>(
                    X[i], WPS[i], SB[i], Y[i], 64, 64, nTiles); break;
            case 256:
                gemm_f32_kernel<256, 64><<<grid, 256, 0, stream>>>(
                    X[i], WPS[i], SB[i], Y[i], 64, 64, nTiles); break;
            default:
                gemm_f32_kernel<512, 64><<<grid, 256, 0, stream>>>(
                    X[i], WPS[i], SB[i], Y[i], 64, 64, nTiles); break;
        }
        colstats_kernel<<<128, 256, 0, stream>>>(Y[i], STATS + i * 128, N, 64);
        bn_finalize_kernel<<<1, 64, 0, stream>>>(STATS + i * 128, SG[i], SBe[i],
                                                 SS + i * 128, N, 64);
        size_t total = (size_t)N * 64;
        bn_relu_kernel<<<(unsigned)((total + 255) / 256), 256, 0, stream>>>(
            Y[i], SS + i * 128, total, 64);
    }

    // ---- kNN argmin for stages 1..3 (queries = p0) ----
    for (int i = 1; i < 4; ++i) {
        pack_cand_kernel<<<(NS[i] + 255) / 256, 256, 0, stream>>>(
            P[i], CLO[i], CHI[i], NS[i]);
        knn_argmin_kernel<<<NS[0] / 256, 256, 0, stream>>>(
            P[0], CLO[i], CHI[i], IDX[i], NS[0], NS[i]);
    }

    // ---- classifier GEMM1 with fused gather; BN; ReLU ----
    {
        int nTiles = (NS[0] / 16) * (128 / 16);
        gemm_cat_kernel<<<(nTiles + 7) / 8, 256, 0, stream>>>(
            Y[0], Y[1], Y[2], Y[3], IDX[1], IDX[2], IDX[3], WP1, b1, H, nTiles);
        colstats_kernel<<<128, 256, 0, stream>>>(H, STATS + 512, NS[0], 128);
        bn_finalize_kernel<<<1, 128, 0, stream>>>(STATS + 512, g1, beta1,
                                                  SS + 512, NS[0], 128);
        size_t total = (size_t)NS[0] * 128;
        bn_relu_kernel<<<(unsigned)((total + 255) / 256), 256, 0, stream>>>(
            H, SS + 512, total, 128);
    }

    // ---- final Linear(128,17): padded pair W2, masked store of 17 cols ----
    {
        int nTiles = (NS[0] / 16) * (32 / 16);
        gemm_f32_kernel<128, 32><<<(nTiles + 7) / 8, 256, 0, stream>>>(
            H, W2P, B2P, out, 17, 17, nTiles);
    }
}